// BPRLoss_2439541424725
// MI455X (gfx1250) — compile-verified
//
#include <hip/hip_runtime.h>

// BPR-style negative-sampling loss, MI455X (gfx1250, wave32).
//
// Roofline: the big [B,T,N] tensor is only *gathered* (B*T*(1+S) scalar
// loads ~ 25KB), so this is latency-bound, not bandwidth/FLOP bound.
// Structure: 1 wave per user b; lanes stride over t; per-lane stable
// log-sigmoid accumulation in f32; wave-level sum done with
// V_WMMA_F32_16X16X4_F32 (full f32 matrix op, no precision loss):
//   A(16x4) holds the 32 lane partials (A-vgpr0=v, A-vgpr1=0),
//   B(4x16) == 1.0  =>  D[m,n] = sum_k A[m,k]
//   column n of D lives in lanes {n, n+16} x 8 VGPRs, so
//   sum(c[0..7]) + shfl_xor(.,16) = full wave sum in every lane.

typedef __attribute__((ext_vector_type(2))) float v2f;
typedef __attribute__((ext_vector_type(8))) float v8f;

__device__ __forceinline__ float wave32_sum_wmma(float v) {
  v2f a;
  a[0] = v;      // this lane's partial -> one A element
  a[1] = 0.0f;   // second A element unused
  v2f b;
  b[0] = 1.0f;   // B == all-ones (layout-agnostic)
  b[1] = 1.0f;
  v8f c = {};    // zero accumulator
  // 8 args: (neg_a, A, neg_b, B, c_mod, C, reuse_a, reuse_b)
  c = __builtin_amdgcn_wmma_f32_16x16x4_f32(
      /*neg_a=*/false, a, /*neg_b=*/false, b,
      /*c_mod=*/(short)0, c, /*reuse_a=*/false, /*reuse_b=*/false);
  float local = c[0] + c[1] + c[2] + c[3] + c[4] + c[5] + c[6] + c[7];
  // rows 0..7 of column n are in lane n, rows 8..15 in lane n+16
  return local + __shfl_xor(local, 16, 32);
}

__device__ __forceinline__ float log_sigmoid_stable(float x) {
  // log(sigmoid(x)) = min(x,0) - log1p(exp(-|x|))
  return fminf(x, 0.0f) - log1pf(__expf(-fabsf(x)));
}

// One block (one wave32) per user b.
__global__ void bpr_per_user_kernel(const float* __restrict__ output,
                                    const int* __restrict__ labels,
                                    const int* __restrict__ x_lens,
                                    const int* __restrict__ neg_ids,
                                    float* __restrict__ per_user,
                                    int T, long long N, int S) {
  const int b = blockIdx.x;
  const int lane = threadIdx.x;
  const int xlen = x_lens[b];
  const float* out_b = output + (long long)b * (long long)T * N;

  float acc = 0.0f;
  for (int t = lane; t < T; t += 32) {
    if (t < xlen) {  // mask: only valid timesteps contribute
      const int lbl = labels[b * T + t];
      const float pos = out_b[(long long)t * N + lbl];
      float s = 0.0f;
      const int base = (b * T + t) * S;
      for (int si = 0; si < S; ++si) {
        const int nid = neg_ids[base + si];
        const float neg = out_b[(long long)t * N + nid];
        s += log_sigmoid_stable(pos - neg);
      }
      acc += s;
    }
  }
  // reconverged here: EXEC all-ones for WMMA
  const float total = wave32_sum_wmma(acc);
  if (lane == 0) per_user[b] = total / (float)xlen;
}

// Single wave: reduce per_user[0..B-1] -> -mean
__global__ void bpr_finalize_kernel(const float* __restrict__ per_user,
                                    float* __restrict__ out, int B) {
  const int lane = threadIdx.x;
  float acc = 0.0f;
  for (int i = lane; i < B; i += 32) acc += per_user[i];
  const float total = wave32_sum_wmma(acc);
  if (lane == 0) out[0] = -total / (float)B;
}

extern "C" void kernel_launch(void* const* d_in, const int* in_sizes, int n_in,
                              void* d_out, int out_size, void* d_ws, size_t ws_size,
                              hipStream_t stream) {
  const float* output  = (const float*)d_in[0];  // [B,T,N] f32
  const int*   labels  = (const int*)d_in[1];    // [B,T]
  const int*   x_lens  = (const int*)d_in[2];    // [B]
  const int*   neg_ids = (const int*)d_in[3];    // [B,T,S]
  // d_in[4] = uids, unused by the reference math

  const int B = in_sizes[2];
  const int T = in_sizes[1] / B;
  const long long N = (long long)in_sizes[0] / ((long long)B * (long long)T);
  const int S = in_sizes[3] / (B * T);

  float* per_user = (float*)d_ws;  // B floats of scratch

  bpr_per_user_kernel<<<B, 32, 0, stream>>>(output, labels, x_lens, neg_ids,
                                            per_user, T, N, S);
  bpr_finalize_kernel<<<1, 32, 0, stream>>>(per_user, (float*)d_out, B);
}